// APPNPNet_36498632082157
// MI455X (gfx1250) — compile-verified
//
#include <hip/hip_runtime.h>
#include <hip/hip_bf16.h>

typedef __attribute__((ext_vector_type(2))) float v2f;
typedef __attribute__((ext_vector_type(8))) float v8f;

#define ALPHA 0.1f

// ---------------------------------------------------------------------------
// init: zero hB (first propagation target) and set deg=1.0 (self-loop)
// ---------------------------------------------------------------------------
__global__ __launch_bounds__(256) void appnp_init_kernel(float* __restrict__ hB,
                                                         float* __restrict__ deg,
                                                         int n) {
    int tid = blockIdx.x * blockDim.x + threadIdx.x;
    int n16 = n * 16;
    if (tid < n16) {
        float4 z; z.x = 0.f; z.y = 0.f; z.z = 0.f; z.w = 0.f;
        *(float4*)(hB + (size_t)tid * 4) = z;
    }
    if (tid < n) deg[tid] = 1.0f;
}

// ---------------------------------------------------------------------------
// degree accumulation over edge targets
// ---------------------------------------------------------------------------
__global__ __launch_bounds__(256) void appnp_degree_kernel(const long long* __restrict__ col,
                                                           float* __restrict__ deg,
                                                           int E) {
    int e = blockIdx.x * blockDim.x + threadIdx.x;
    if (e < E) {
        atomicAdd(&deg[(int)col[e]], 1.0f);
    }
}

__global__ __launch_bounds__(256) void appnp_dinv_kernel(const float* __restrict__ deg,
                                                         float* __restrict__ dinv,
                                                         int n) {
    int i = blockIdx.x * blockDim.x + threadIdx.x;
    if (i < n) dinv[i] = __frsqrt_rn(deg[i]);   // deg >= 1 always (self loop)
}

// ---------------------------------------------------------------------------
// MLP encoder with f32 WMMA:  h0 = relu(x@W1 + b1) @ W2 + b2
//   256 threads = 8 waves; each wave computes a 16-row tile.
//   GEMM1: [16,512]x[512,16] -> 128 x v_wmma_f32_16x16x4_f32
//   GEMM2: [16,16]x[16,64]   -> 16  x v_wmma_f32_16x16x4_f32 (A restaged via LDS)
// ---------------------------------------------------------------------------
__global__ __launch_bounds__(256) void appnp_mlp_kernel(const float* __restrict__ x,
                                                        const float* __restrict__ W1,
                                                        const float* __restrict__ b1,
                                                        const float* __restrict__ W2,
                                                        const float* __restrict__ b2,
                                                        float* __restrict__ h0,
                                                        float* __restrict__ hA,
                                                        int n) {
    __shared__ float sW1[512 * 16];     // 32 KB
    __shared__ float sW2[16 * 64];      //  4 KB
    __shared__ float sH1[8][16 * 16];   //  8 KB (per-wave relu(h1) tile)

    int t = threadIdx.x;
    for (int i = t; i < 512 * 16; i += 256) sW1[i] = W1[i];
    for (int i = t; i < 16 * 64; i += 256) sW2[i] = W2[i];
    __syncthreads();

    int wave = t >> 5;
    int lane = t & 31;
    int m    = lane & 15;            // row index of A / col index of B,C,D
    int kh   = (lane >> 4) << 1;     // K sub-offset: 0 (lanes 0-15) or 2 (lanes 16-31)
    int mb   = (lane >> 4) << 3;     // C/D row base: 0 or 8

    int rowBase = (blockIdx.x * 8 + wave) * 16;
    int r = rowBase + m;
    if (r >= n) r = n - 1;           // clamp loads; stores are guarded
    const float* xrow = x + (size_t)r * 512;

    // ---- GEMM1: acc[16x16] = x_tile[16x512] @ W1[512x16] ----
    v8f acc = {};
#pragma unroll 4
    for (int k0 = 0; k0 < 512; k0 += 4) {
        v2f a, b;
        a.x = xrow[k0 + kh];
        a.y = xrow[k0 + kh + 1];
        b.x = sW1[(k0 + kh) * 16 + m];
        b.y = sW1[(k0 + kh + 1) * 16 + m];
        acc = __builtin_amdgcn_wmma_f32_16x16x4_f32(false, a, false, b,
                                                    (short)0, acc, false, false);
    }

    // bias + relu, stage tile in LDS to re-form the A layout (wave-local)
    float bias1 = b1[m];
    float* ht = &sH1[wave][0];
#pragma unroll
    for (int v = 0; v < 8; ++v) {
        float val = acc[v] + bias1;
        val = val > 0.0f ? val : 0.0f;
        ht[(mb + v) * 16 + m] = val;
    }

    // ---- GEMM2: out[16x64] = h1[16x16] @ W2[16x64], 4 column tiles ----
#pragma unroll
    for (int nt = 0; nt < 4; ++nt) {
        v8f acc2 = {};
#pragma unroll
        for (int k0 = 0; k0 < 16; k0 += 4) {
            v2f a, b;
            a.x = ht[m * 16 + k0 + kh];
            a.y = ht[m * 16 + k0 + kh + 1];
            b.x = sW2[(k0 + kh) * 64 + nt * 16 + m];
            b.y = sW2[(k0 + kh + 1) * 64 + nt * 16 + m];
            acc2 = __builtin_amdgcn_wmma_f32_16x16x4_f32(false, a, false, b,
                                                         (short)0, acc2, false, false);
        }
        float bias2 = b2[nt * 16 + m];
#pragma unroll
        for (int v = 0; v < 8; ++v) {
            int rr = rowBase + mb + v;
            if (rr < n) {
                float val = acc2[v] + bias2;
                size_t idx = (size_t)rr * 64 + nt * 16 + m;
                h0[idx] = val;
                hA[idx] = val;
            }
        }
    }
}

// ---------------------------------------------------------------------------
// scatter: agg[col] += dinv[row]*dinv[col] * h[row]   (16 threads per edge)
// ---------------------------------------------------------------------------
__global__ __launch_bounds__(256) void appnp_scatter_kernel(const float* __restrict__ h,
                                                            const float* __restrict__ dinv,
                                                            const long long* __restrict__ row,
                                                            const long long* __restrict__ col,
                                                            float* __restrict__ agg,
                                                            int E) {
    long long tid = (long long)blockIdx.x * blockDim.x + threadIdx.x;
    int e = (int)(tid >> 4);
    if (e >= E) return;
    int c4 = ((int)tid & 15) << 2;

    int rsrc = (int)row[e];
    int rdst = (int)col[e];
    float nrm = dinv[rsrc] * dinv[rdst];

    float4 hv = *(const float4*)(h + (size_t)rsrc * 64 + c4);
    float* ap = agg + (size_t)rdst * 64 + c4;
    atomicAdd(ap + 0, nrm * hv.x);
    atomicAdd(ap + 1, nrm * hv.y);
    atomicAdd(ap + 2, nrm * hv.z);
    atomicAdd(ap + 3, nrm * hv.w);
}

// ---------------------------------------------------------------------------
// combine (in place): agg <- 0.9*(agg + dinv^2 * hsrc) + 0.1*h0 ; hsrc <- 0
//   the zeroed hsrc becomes the scatter target of the next step
// ---------------------------------------------------------------------------
__global__ __launch_bounds__(256) void appnp_combine_kernel(float* __restrict__ agg,
                                                            float* __restrict__ hsrc,
                                                            const float* __restrict__ h0,
                                                            const float* __restrict__ dinv,
                                                            int n) {
    int tid = blockIdx.x * blockDim.x + threadIdx.x;
    if (tid >= n * 16) return;
    int node = tid >> 4;
    float di = dinv[node];
    float sl = di * di;
    size_t base = (size_t)tid * 4;

    float4 a  = *(float4*)(agg + base);
    float4 hs = *(float4*)(hsrc + base);
    float4 h0v = *(const float4*)(h0 + base);
    float4 o;
    o.x = (1.0f - ALPHA) * (a.x + sl * hs.x) + ALPHA * h0v.x;
    o.y = (1.0f - ALPHA) * (a.y + sl * hs.y) + ALPHA * h0v.y;
    o.z = (1.0f - ALPHA) * (a.z + sl * hs.z) + ALPHA * h0v.z;
    o.w = (1.0f - ALPHA) * (a.w + sl * hs.w) + ALPHA * h0v.w;
    *(float4*)(agg + base) = o;

    float4 z; z.x = 0.f; z.y = 0.f; z.z = 0.f; z.w = 0.f;
    *(float4*)(hsrc + base) = z;
}

// ---------------------------------------------------------------------------
// log-softmax over 64 channels: one wave32 per node, 2 channels per lane
// ---------------------------------------------------------------------------
__global__ __launch_bounds__(256) void appnp_lsm_kernel(const float* __restrict__ h,
                                                        float* __restrict__ out,
                                                        int n) {
    int wave = threadIdx.x >> 5;
    int lane = threadIdx.x & 31;
    int node = blockIdx.x * 8 + wave;
    if (node >= n) return;

    const float* hp = h + (size_t)node * 64;
    float v0 = hp[lane];
    float v1 = hp[lane + 32];

    float mx = fmaxf(v0, v1);
#pragma unroll
    for (int off = 16; off >= 1; off >>= 1)
        mx = fmaxf(mx, __shfl_xor(mx, off, 32));

    float s = __expf(v0 - mx) + __expf(v1 - mx);
#pragma unroll
    for (int off = 16; off >= 1; off >>= 1)
        s += __shfl_xor(s, off, 32);

    float lg = __logf(s);
    float* op = out + (size_t)node * 64;
    op[lane]      = v0 - mx - lg;
    op[lane + 32] = v1 - mx - lg;
}

// ---------------------------------------------------------------------------
// launch
// ---------------------------------------------------------------------------
extern "C" void kernel_launch(void* const* d_in, const int* in_sizes, int n_in,
                              void* d_out, int out_size, void* d_ws, size_t ws_size,
                              hipStream_t stream) {
    const float* x  = (const float*)d_in[0];
    const float* W1 = (const float*)d_in[1];
    const float* b1 = (const float*)d_in[2];
    const float* W2 = (const float*)d_in[3];
    const float* b2 = (const float*)d_in[4];
    const long long* ei = (const long long*)d_in[5];

    int n = in_sizes[0] / 512;         // 100000
    int E = in_sizes[5] / 2;           // 3200000
    const long long* row = ei;         // edge_index[0] = sources
    const long long* col = ei + E;     // edge_index[1] = targets

    float* ws   = (float*)d_ws;
    float* h0   = ws;
    float* hA   = h0 + (size_t)n * 64;
    float* hB   = hA + (size_t)n * 64;
    float* deg  = hB + (size_t)n * 64;
    float* dinv = deg + n;
    float* out  = (float*)d_out;

    int n16 = n * 16;

    appnp_init_kernel<<<(n16 + 255) / 256, 256, 0, stream>>>(hB, deg, n);
    appnp_degree_kernel<<<(E + 255) / 256, 256, 0, stream>>>(col, deg, E);
    appnp_dinv_kernel<<<(n + 255) / 256, 256, 0, stream>>>(deg, dinv, n);

    appnp_mlp_kernel<<<(n + 127) / 128, 256, 0, stream>>>(x, W1, b1, W2, b2, h0, hA, n);

    float* hsrc = hA;
    float* hdst = hB;
    long long scatter_threads = (long long)E * 16;
    int scatter_blocks = (int)((scatter_threads + 255) / 256);
    for (int k = 0; k < 10; ++k) {
        appnp_scatter_kernel<<<scatter_blocks, 256, 0, stream>>>(hsrc, dinv, row, col, hdst, E);
        appnp_combine_kernel<<<(n16 + 255) / 256, 256, 0, stream>>>(hdst, hsrc, h0, dinv, n);
        float* tmp = hsrc; hsrc = hdst; hdst = tmp;
    }

    appnp_lsm_kernel<<<(n + 7) / 8, 256, 0, stream>>>(hsrc, out, n);
}